// RecurrentEntityNetwork_32890859552825
// MI455X (gfx1250) — compile-verified
//
#include <hip/hip_runtime.h>
#include <hip/hip_bf16.h>

// Problem constants (from reference): B=512, T=1005, E=100, L=10, N=20, V=32000
#define B_DIM 512
#define T_DIM 1005
#define E_DIM 100
#define L_DIM 10
#define N_ENT 20
#define V_DIM 32000
#define NF    100   // T / L
#define REST  5     // T % L

typedef float v2f __attribute__((ext_vector_type(2)));
typedef float v8f __attribute__((ext_vector_type(8)));

// ---------------------------------------------------------------------------
// Kernel 1: infos[b,n,e] = sum_l f[l,e]*in_data[b, n*L+l, e]  (n < NF)
//           q[b,e]      = sum_l f[l,e]*in_data[b, NF*L+l, e]  (block n == NF)
// grid (NF+1, B), block 128
// ---------------------------------------------------------------------------
__global__ void ren_infos_q(const float* __restrict__ in_data,
                            const float* __restrict__ f,
                            float* __restrict__ infos,
                            float* __restrict__ qout)
{
    const int b = blockIdx.y;
    const int n = blockIdx.x;
    const int e = threadIdx.x;
    if (e >= E_DIM) return;
    if (n < NF) {
        const float* base = in_data + ((size_t)b * T_DIM + (size_t)n * L_DIM) * E_DIM + e;
        float s = 0.f;
        #pragma unroll
        for (int l = 0; l < L_DIM; ++l)
            s += f[l * E_DIM + e] * base[(size_t)l * E_DIM];
        infos[((size_t)b * NF + n) * E_DIM + e] = s;
    } else {
        const float* base = in_data + ((size_t)b * T_DIM + (size_t)NF * L_DIM) * E_DIM + e;
        float s = 0.f;
        #pragma unroll
        for (int l = 0; l < REST; ++l)
            s += f[l * E_DIM + e] * base[(size_t)l * E_DIM];
        qout[(size_t)b * E_DIM + e] = s;
    }
}

// ---------------------------------------------------------------------------
// Kernel 2: Vw[n,f] = sum_e w_mem[n,e] * V_w[f,e]   (20x100, tiny)
// grid N_ENT, block 128
// ---------------------------------------------------------------------------
__global__ void ren_vw(const float* __restrict__ w_mem,
                       const float* __restrict__ V_w,
                       float* __restrict__ Vw)
{
    const int n = blockIdx.x;
    const int fd = threadIdx.x;
    if (fd >= E_DIM) return;
    float s = 0.f;
    for (int e = 0; e < E_DIM; ++e)
        s += w_mem[n * E_DIM + e] * V_w[fd * E_DIM + e];
    Vw[n * E_DIM + fd] = s;
}

// ---------------------------------------------------------------------------
// Generic WMMA f32 GEMM:  C[M,N] = A[M,K] @ B'  (+bias),  B'(k,n)=Bsrc[n*ldb+k]
// K = 4*KSTEPS at compile time (no K guards -> unconditional b64 frag loads).
// Each wave owns TWO 16x16 tiles stacked in M sharing one B fragment.
// grid (ceil(N/128), M/32), block 256 (8 waves). M must be a multiple of 32.
// lda/ldb must be even (8B-aligned float2 fragment loads).
// A frag layout: lanes 0-15 -> M=lane, K={k0,k0+1}; lanes 16-31 -> K={k0+2,k0+3}.
// B frag layout (mirror): lanes 0-15 -> N=lane, same K split.
// ---------------------------------------------------------------------------
template<int KSTEPS>
__global__ void ren_wmma_gemm_nt(const float* __restrict__ A,
                                 const float* __restrict__ Bsrc,
                                 const float* __restrict__ bias,
                                 float* __restrict__ C,
                                 int N, int lda, int ldb, int ldc)
{
    const int wave  = threadIdx.x >> 5;     // 0..7
    const int lane  = threadIdx.x & 31;
    const int l16   = lane & 15;
    const int khalf = (lane >> 4) * 2;      // 0 or 2
    const int mbase = blockIdx.y * 32;
    const int n     = blockIdx.x * 128 + wave * 16 + l16;
    const int nc    = (n < N) ? n : (N - 1);   // branchless clamp, EXEC untouched

    const float* Arow0 = A    + (size_t)(mbase + l16) * lda + khalf;
    const float* Arow1 = Arow0 + (size_t)16 * lda;
    const float* Brow  = Bsrc + (size_t)nc * ldb + khalf;

    v8f acc0 = {}, acc1 = {};
    #pragma unroll
    for (int kk = 0; kk < KSTEPS; ++kk) {
        const v2f a0 = *(const v2f*)(Arow0 + kk * 4);
        const v2f a1 = *(const v2f*)(Arow1 + kk * 4);
        const v2f b  = *(const v2f*)(Brow  + kk * 4);
        acc0 = __builtin_amdgcn_wmma_f32_16x16x4_f32(false, a0, false, b,
                                                     (short)0, acc0, false, false);
        acc1 = __builtin_amdgcn_wmma_f32_16x16x4_f32(false, a1, false, b,
                                                     (short)0, acc1, false, false);
    }

    if (n < N) {
        const float bv = bias ? bias[n] : 0.f;
        const int r0 = mbase + ((lane >> 4) << 3);
        #pragma unroll
        for (int r = 0; r < 8; ++r) {
            C[(size_t)(r0 + r)      * ldc + n] = acc0[r] + bv;
            C[(size_t)(r0 + 16 + r) * ldc + n] = acc1[r] + bv;
        }
    }
}

// ---------------------------------------------------------------------------
// Kernel 4: per-batch recurrent scan (100 sequential steps), 1 WG per batch.
// State h (20x100) + U_w^T (100x112) live in LDS for the whole scan.
// cand = prelu(h @ U_w^T + Vw + sW, a_dm) via WMMA f32 16x16x4 tiles;
// sW was precomputed by the hoisted GEMM (sw_all).
// grid B, block 256 (8 waves), dynamic LDS (~92 KB; WGP has 320 KB).
// ---------------------------------------------------------------------------
#define UW_STR   112   // padded N dim (7 tiles of 16)
#define H_STR    104
#define C_STR    120

#define LDS_UWT   (100 * UW_STR)          // 11200
#define LDS_HBUF  (32 * H_STR)            //  3328
#define LDS_WM    (N_ENT * H_STR)         //  2080
#define LDS_VWL   (N_ENT * H_STR)         //  2080
#define LDS_CAND  (32 * C_STR)            //  3840
#define LDS_SVEC  104
#define LDS_SWV   112
#define LDS_GV    32
#define LDS_PART  256
#define LDS_NRM   32
#define SCAN_LDS_FLOATS (LDS_UWT + LDS_HBUF + LDS_WM + LDS_VWL + LDS_CAND + \
                         LDS_SVEC + LDS_SWV + LDS_GV + LDS_PART + LDS_NRM)

__global__ void ren_scan(const float* __restrict__ infos,
                         const float* __restrict__ sw_all,
                         const float* __restrict__ h0,
                         const float* __restrict__ w_mem,
                         const float* __restrict__ U_w,
                         const float* __restrict__ Vw,
                         const float* __restrict__ a_dm,
                         float* __restrict__ h_out)
{
    extern __shared__ float lds[];
    float* UwT  = lds;                        // [100][UW_STR]  B-matrix (k,f)
    float* hbuf = UwT  + LDS_UWT;             // [32][H_STR]    A-matrix (padded)
    float* wm   = hbuf + LDS_HBUF;            // [20][H_STR]
    float* Vwl  = wm   + LDS_WM;              // [20][H_STR]
    float* cand = Vwl  + LDS_VWL;             // [32][C_STR]
    float* svec = cand + LDS_CAND;            // [104]
    float* swv  = svec + LDS_SVEC;            // [112]
    float* gv   = swv  + LDS_SWV;             // [32]
    float* part = gv   + LDS_GV;              // [256]
    float* nrm  = part + LDS_PART;            // [32]

    const int tid = threadIdx.x;
    const int b   = blockIdx.x;
    const float adm = a_dm[0];

    // Load U_w transposed: UwT[k][f] = U_w[f*E + k], zero-pad f in [100,112)
    for (int i = tid; i < LDS_UWT; i += 256) {
        const int k = i / UW_STR, fd = i % UW_STR;
        UwT[i] = (fd < E_DIM) ? U_w[fd * E_DIM + k] : 0.f;
    }
    // h init (rows >= 20 and cols >= 100 stay zero for WMMA padding)
    for (int i = tid; i < LDS_HBUF; i += 256) {
        const int n = i / H_STR, e = i % H_STR;
        hbuf[i] = (n < N_ENT && e < E_DIM) ? h0[n * E_DIM + e] : 0.f;
    }
    for (int i = tid; i < LDS_WM; i += 256) {
        const int n = i / H_STR, e = i % H_STR;
        wm[i]  = (e < E_DIM) ? w_mem[n * E_DIM + e] : 0.f;
        Vwl[i] = (e < E_DIM) ? Vw[n * E_DIM + e]    : 0.f;
    }
    __syncthreads();

    const int wave = tid >> 5, lane = tid & 31;
    const int l16 = lane & 15, khalf = (lane >> 4) * 2;
    const int nrow = tid >> 3, sub = tid & 7;   // 8 threads per entity row

    for (int t = 0; t < NF; ++t) {
        const float* srow  = infos  + ((size_t)b * NF + t) * E_DIM;
        const float* swrow = sw_all + ((size_t)b * NF + t) * E_DIM;
        if (tid < 112) {
            if (tid < 104) svec[tid] = (tid < E_DIM) ? srow[tid] : 0.f;
            swv[tid] = (tid < E_DIM) ? swrow[tid] : 0.f;
        }
        __syncthreads();

        // gate: g[n] = sigmoid( sum_e (h[n,e]+wm[n,e]) * s[e] )
        if (nrow < N_ENT) {
            float p = 0.f;
            for (int e = sub; e < E_DIM; e += 8)
                p += (hbuf[nrow * H_STR + e] + wm[nrow * H_STR + e]) * svec[e];
            part[tid] = p;
        }
        __syncthreads();
        if (tid < N_ENT) {
            float d = 0.f;
            #pragma unroll
            for (int j = 0; j < 8; ++j) d += part[tid * 8 + j];
            gv[tid] = 1.f / (1.f + __expf(-d));
        }
        __syncthreads();

        // cand = prelu(h @ UwT + Vw + sW): 2x7 = 14 WMMA 16x16 tiles
        for (int tt = wave; tt < 14; tt += 8) {       // wave-uniform loop
            const int mt = tt / 7, nt = tt % 7;
            const float* arow = hbuf + (size_t)(mt * 16 + l16) * H_STR + khalf;
            const float* bcol = UwT + nt * 16 + l16;
            v8f acc = {};
            #pragma unroll 5
            for (int kk = 0; kk < 25; ++kk) {
                const int k0 = kk * 4 + khalf;
                const v2f a = *(const v2f*)(arow + kk * 4);   // ds_load_b64
                v2f bf;
                bf.x = bcol[(size_t)k0 * UW_STR];
                bf.y = bcol[(size_t)(k0 + 1) * UW_STR];
                acc = __builtin_amdgcn_wmma_f32_16x16x4_f32(false, a, false, bf,
                                                            (short)0, acc, false, false);
            }
            // Branchless epilogue: clamp read indices (stay in-bounds), write
            // unconditionally. Garbage lands only in cand padding (rows>=20 or
            // cols>=100), which the update phase never reads. EXEC untouched.
            const int col = nt * 16 + l16;
            const int ccl = (col < E_DIM) ? col : (E_DIM - 1);
            const int rb  = mt * 16 + ((lane >> 4) << 3);
            const float sws = swv[ccl];
            #pragma unroll
            for (int r = 0; r < 8; ++r) {
                const int row = rb + r;
                const int rcl = (row < N_ENT) ? row : (N_ENT - 1);
                float v = acc[r] + Vwl[rcl * H_STR + ccl] + sws;
                v = (v >= 0.f) ? v : adm * v;     // prelu
                cand[row * C_STR + col] = v;
            }
        }
        __syncthreads();

        // h += g*cand ; accumulate squared norm
        if (nrow < N_ENT) {
            const float gn = gv[nrow];
            float ss = 0.f;
            for (int e = sub; e < E_DIM; e += 8) {
                const float hn = hbuf[nrow * H_STR + e] + gn * cand[nrow * C_STR + e];
                hbuf[nrow * H_STR + e] = hn;
                ss += hn * hn;
            }
            part[tid] = ss;
        }
        __syncthreads();
        if (tid < N_ENT) {
            float ssum = 0.f;
            #pragma unroll
            for (int j = 0; j < 8; ++j) ssum += part[tid * 8 + j];
            nrm[tid] = rsqrtf(ssum);
        }
        __syncthreads();
        if (nrow < N_ENT) {
            const float r = nrm[nrow];
            for (int e = sub; e < E_DIM; e += 8)
                hbuf[nrow * H_STR + e] *= r;
        }
        __syncthreads();
    }

    for (int i = tid; i < N_ENT * E_DIM; i += 256) {
        const int n = i / E_DIM, e = i % E_DIM;
        h_out[((size_t)b * N_ENT + n) * E_DIM + e] = hbuf[n * H_STR + e];
    }
}

// ---------------------------------------------------------------------------
// Kernel 5: attention head per batch: p = softmax(h.q), u = p.h,
// y = q + u @ H_w^T + H_b, yact = prelu(y, a_out).  grid B, block 128.
// ---------------------------------------------------------------------------
__global__ void ren_head(const float* __restrict__ h_final,
                         const float* __restrict__ qbuf,
                         const float* __restrict__ H_w,
                         const float* __restrict__ H_b,
                         const float* __restrict__ a_out,
                         float* __restrict__ yact)
{
    __shared__ float hq[N_ENT];
    __shared__ float p[N_ENT];
    __shared__ float u[E_DIM];
    __shared__ float qv[E_DIM];
    __shared__ float invs;
    const int b = blockIdx.x;
    const int tid = threadIdx.x;
    const float* h = h_final + (size_t)b * N_ENT * E_DIM;

    if (tid < E_DIM) qv[tid] = qbuf[(size_t)b * E_DIM + tid];
    __syncthreads();
    if (tid < N_ENT) {
        float d = 0.f;
        for (int e = 0; e < E_DIM; ++e) d += h[tid * E_DIM + e] * qv[e];
        hq[tid] = d;
    }
    __syncthreads();
    if (tid == 0) {
        float m = hq[0];
        for (int n = 1; n < N_ENT; ++n) m = fmaxf(m, hq[n]);
        float s = 0.f;
        for (int n = 0; n < N_ENT; ++n) { p[n] = __expf(hq[n] - m); s += p[n]; }
        invs = 1.f / s;
    }
    __syncthreads();
    if (tid < E_DIM) {
        float s = 0.f;
        const float inv = invs;
        for (int n = 0; n < N_ENT; ++n) s += p[n] * inv * h[n * E_DIM + tid];
        u[tid] = s;
    }
    __syncthreads();
    if (tid < E_DIM) {
        float y = qv[tid] + H_b[tid];
        for (int e = 0; e < E_DIM; ++e) y += u[e] * H_w[tid * E_DIM + e];
        const float ao = a_out[0];
        yact[(size_t)b * E_DIM + tid] = (y >= 0.f) ? y : ao * y;
    }
}

// ---------------------------------------------------------------------------
// Launch
// ---------------------------------------------------------------------------
extern "C" void kernel_launch(void* const* d_in, const int* in_sizes, int n_in,
                              void* d_out, int out_size, void* d_ws, size_t ws_size,
                              hipStream_t stream)
{
    const float* in_data = (const float*)d_in[0];
    const float* f       = (const float*)d_in[1];
    const float* h0      = (const float*)d_in[2];
    const float* w_mem   = (const float*)d_in[3];
    const float* U_w     = (const float*)d_in[4];
    const float* V_w     = (const float*)d_in[5];
    const float* W_w     = (const float*)d_in[6];
    const float* a_dm    = (const float*)d_in[7];
    const float* H_w     = (const float*)d_in[8];
    const float* H_b     = (const float*)d_in[9];
    const float* R_w     = (const float*)d_in[10];
    const float* R_b     = (const float*)d_in[11];
    const float* a_out   = (const float*)d_in[12];
    float* out = (float*)d_out;

    // workspace layout (floats)
    float* ws = (float*)d_ws;
    float* infos  = ws;                                   // B*NF*E   = 5,120,000
    float* qbuf   = infos  + (size_t)B_DIM * NF * E_DIM;  // B*E      =    51,200
    float* Vw     = qbuf   + (size_t)B_DIM * E_DIM;       // N*E      =     2,000
    float* sw_all = Vw     + (size_t)N_ENT * E_DIM;       // B*NF*E   = 5,120,000
    float* h_fin  = sw_all + (size_t)B_DIM * NF * E_DIM;  // B*N*E    = 1,024,000
    float* yact   = h_fin  + (size_t)B_DIM * N_ENT * E_DIM; // B*E    =    51,200
    (void)ws_size; (void)n_in; (void)in_sizes; (void)out_size;

    // 1) chunk summaries + query
    ren_infos_q<<<dim3(NF + 1, B_DIM), 128, 0, stream>>>(in_data, f, infos, qbuf);

    // 2) Vw = w_mem @ V_w^T
    ren_vw<<<N_ENT, 128, 0, stream>>>(w_mem, V_w, Vw);

    // 3) hoisted sW GEMM: sw_all[(b,t),f] = infos[(b,t),:] @ W_w^T
    //    M=51200, N=100, K=100 (KSTEPS=25)
    ren_wmma_gemm_nt<25><<<dim3(1, (B_DIM * NF) / 32), 256, 0, stream>>>(
        infos, W_w, nullptr, sw_all,
        E_DIM, E_DIM, E_DIM, E_DIM);

    // 4) recurrent scan (one WG per batch, state + U_w^T resident in LDS)
    ren_scan<<<B_DIM, 256, SCAN_LDS_FLOATS * sizeof(float), stream>>>(
        infos, sw_all, h0, w_mem, U_w, Vw, a_dm, h_fin);

    // 5) attention head -> activated y
    ren_head<<<B_DIM, 128, 0, stream>>>(h_fin, qbuf, H_w, H_b, a_out, yact);

    // 6) output projection: out = yact @ R_w^T + R_b  (M=512, N=32000, K=100)
    ren_wmma_gemm_nt<25><<<dim3((V_DIM + 127) / 128, B_DIM / 32), 256, 0, stream>>>(
        yact, R_w, R_b, out,
        V_DIM, E_DIM, E_DIM, V_DIM);
}